// FreqDecompMoE_34394098106685
// MI455X (gfx1250) — compile-verified
//
#include <hip/hip_runtime.h>
#include <math.h>

// Problem constants (from reference)
#define NB 32
#define NC 512
#define NL 2048
#define NF 1025
#define NE 8

typedef float v8f __attribute__((ext_vector_type(8)));
typedef float v2f __attribute__((ext_vector_type(2)));

#define PI_F 3.14159265358979323846f

// D = A(16x4 f32) * B(4x16 f32) + C(16x16 f32)  -- CDNA5 WMMA
#define WMMA4(a, b, c) \
  __builtin_amdgcn_wmma_f32_16x16x4_f32(false, (a), false, (b), (short)0, (c), false, false)

// ---------------------------------------------------------------------------
// DFT16 tables in LDS: Wr[k*16+m]=cos(-2pi km/16), Wi=sin(...), Wn=-Wi
// ---------------------------------------------------------------------------
__device__ __forceinline__ void dft_tables_init(float* Wr, float* Wi, float* Wn, int tid)
{
  const int k = tid >> 4, m = tid & 15;
  const int r = (k * m) & 15;                 // W16 periodic mod 16
  float sn, cs;
  __sincosf(-(2.0f * PI_F / 16.0f) * (float)r, &sn, &cs);
  Wr[tid] = cs; Wi[tid] = sn; Wn[tid] = -sn;
}

// ---------------------------------------------------------------------------
// In-LDS forward 2048-point complex FFT, natural-order output.
// Decomposition: 2048 = 8 (per-thread radix-8, DIF) x 16 x 16 (two WMMA
// radix-16 stages).  X[j + 8*k1 + 128*k2].
// Block = 256 threads = 8 waves; wave w owns one 16x16 tile per stage.
// ---------------------------------------------------------------------------
__device__ __forceinline__ void fft2048(float* re, float* im,
                                        const float* Wr, const float* Wi,
                                        const float* Wn, int tid)
{
  const int lane = tid & 31;
  const int wave = tid >> 5;
  const int half = lane >> 4;
  const int lc   = lane & 15;
  const float S  = 0.70710678118654752440f;

  // ---------------- radix-8 (per-thread column t = tid), DIF ----------------
  float zr[8], zi[8];
  #pragma unroll
  for (int n1 = 0; n1 < 8; ++n1) { zr[n1] = re[n1 * 256 + tid]; zi[n1] = im[n1 * 256 + tid]; }

  float br, bi;
  // span 4, twiddles W8^{0..3} = 1, (S,-S), (0,-1), (-S,-S)
  br = zr[0]-zr[4]; bi = zi[0]-zi[4]; zr[0]+=zr[4]; zi[0]+=zi[4]; zr[4]=br;        zi[4]=bi;
  br = zr[1]-zr[5]; bi = zi[1]-zi[5]; zr[1]+=zr[5]; zi[1]+=zi[5]; zr[5]=S*(br+bi); zi[5]=S*(bi-br);
  br = zr[2]-zr[6]; bi = zi[2]-zi[6]; zr[2]+=zr[6]; zi[2]+=zi[6]; zr[6]=bi;        zi[6]=-br;
  br = zr[3]-zr[7]; bi = zi[3]-zi[7]; zr[3]+=zr[7]; zi[3]+=zi[7]; zr[7]=S*(bi-br); zi[7]=-S*(br+bi);
  // span 2, twiddles W4^{0,1} = 1, -i
  br = zr[0]-zr[2]; bi = zi[0]-zi[2]; zr[0]+=zr[2]; zi[0]+=zi[2]; zr[2]=br; zi[2]=bi;
  br = zr[1]-zr[3]; bi = zi[1]-zi[3]; zr[1]+=zr[3]; zi[1]+=zi[3]; zr[3]=bi; zi[3]=-br;
  br = zr[4]-zr[6]; bi = zi[4]-zi[6]; zr[4]+=zr[6]; zi[4]+=zi[6]; zr[6]=br; zi[6]=bi;
  br = zr[5]-zr[7]; bi = zi[5]-zi[7]; zr[5]+=zr[7]; zi[5]+=zi[7]; zr[7]=bi; zi[7]=-br;
  // span 1
  br = zr[0]-zr[1]; bi = zi[0]-zi[1]; zr[0]+=zr[1]; zi[0]+=zi[1]; zr[1]=br; zi[1]=bi;
  br = zr[2]-zr[3]; bi = zi[2]-zi[3]; zr[2]+=zr[3]; zi[2]+=zi[3]; zr[3]=br; zi[3]=bi;
  br = zr[4]-zr[5]; bi = zi[4]-zi[5]; zr[4]+=zr[5]; zi[4]+=zi[5]; zr[5]=br; zi[5]=bi;
  br = zr[6]-zr[7]; bi = zi[6]-zi[7]; zr[6]+=zr[7]; zi[6]+=zi[7]; zr[7]=br; zi[7]=bi;

  __syncthreads();   // time-domain reads complete before overwriting LDS

  // bit-reversed output order; twiddle W_2048^{j*t}; store a[j*256 + t]
  const int rev[8] = {0, 4, 2, 6, 1, 5, 3, 7};
  #pragma unroll
  for (int j = 0; j < 8; ++j) {
    float xr = zr[rev[j]], xi = zi[rev[j]];
    float sn, cs;
    __sincosf(-(2.0f * PI_F / 2048.0f) * (float)(j * tid), &sn, &cs);
    re[j * 256 + tid] = xr * cs - xi * sn;
    im[j * 256 + tid] = xr * sn + xi * cs;
  }
  __syncthreads();

  // ------------- stage 2: DFT16 over m1 (WMMA), wave handles j = wave -------
  // b0[k1,m2] = sum_m1 W16[k1,m1] * a[j, m1*16+m2]
  v8f ar = {0,0,0,0,0,0,0,0}, ai = {0,0,0,0,0,0,0,0};
  const int base = wave * 256;
  #pragma unroll
  for (int kk = 0; kk < 4; ++kk) {
    const int k0 = 4 * kk + 2 * half;            // K indices k0, k0+1 per lane
    v2f wr, wi, wn, bre, bim;
    wr.x = Wr[lc * 16 + k0]; wr.y = Wr[lc * 16 + k0 + 1];
    wi.x = Wi[lc * 16 + k0]; wi.y = Wi[lc * 16 + k0 + 1];
    wn.x = Wn[lc * 16 + k0]; wn.y = Wn[lc * 16 + k0 + 1];
    bre.x = re[base + k0 * 16 + lc]; bre.y = re[base + (k0 + 1) * 16 + lc];
    bim.x = im[base + k0 * 16 + lc]; bim.y = im[base + (k0 + 1) * 16 + lc];
    ar = WMMA4(wr, bre, ar);  ar = WMMA4(wn, bim, ar);   // Re: Wr*Br - Wi*Bi
    ai = WMMA4(wr, bim, ai);  ai = WMMA4(wi, bre, ai);   // Im: Wr*Bi + Wi*Br
  }
  __syncthreads();

  // twiddle W_256^{k1*m2}; store transposed b[m2*128 + j*16 + k1]
  #pragma unroll
  for (int v = 0; v < 8; ++v) {
    const int k1 = v + 8 * half;                 // C/D layout: M = v + 8*(lane/16)
    float sn, cs;
    __sincosf(-(2.0f * PI_F / 256.0f) * (float)(k1 * lc), &sn, &cs);
    float xr = ar[v], xi = ai[v];
    re[lc * 128 + wave * 16 + k1] = xr * cs - xi * sn;
    im[lc * 128 + wave * 16 + k1] = xr * sn + xi * cs;
  }
  __syncthreads();

  // ------------- stage 3: DFT16 over m2 (WMMA); cols (j=wave, k1=lc) --------
  v8f cr = {0,0,0,0,0,0,0,0}, ci = {0,0,0,0,0,0,0,0};
  #pragma unroll
  for (int kk = 0; kk < 4; ++kk) {
    const int k0 = 4 * kk + 2 * half;
    v2f wr, wi, wn, bre, bim;
    wr.x = Wr[lc * 16 + k0]; wr.y = Wr[lc * 16 + k0 + 1];
    wi.x = Wi[lc * 16 + k0]; wi.y = Wi[lc * 16 + k0 + 1];
    wn.x = Wn[lc * 16 + k0]; wn.y = Wn[lc * 16 + k0 + 1];
    bre.x = re[k0 * 128 + wave * 16 + lc]; bre.y = re[(k0 + 1) * 128 + wave * 16 + lc];
    bim.x = im[k0 * 128 + wave * 16 + lc]; bim.y = im[(k0 + 1) * 128 + wave * 16 + lc];
    cr = WMMA4(wr, bre, cr);  cr = WMMA4(wn, bim, cr);
    ci = WMMA4(wr, bim, ci);  ci = WMMA4(wi, bre, ci);
  }
  __syncthreads();

  // natural-order store: X[j + 8*k1 + 128*k2]
  #pragma unroll
  for (int v = 0; v < 8; ++v) {
    const int k2 = v + 8 * half;
    const int k  = wave + 8 * lc + 128 * k2;
    re[k] = cr[v]; im[k] = ci[v];
  }
  __syncthreads();
}

// ---------------------------------------------------------------------------
// Pass A: per-(b,l) mean / unbiased var over channels
// ---------------------------------------------------------------------------
__global__ __launch_bounds__(256)
void fdm_stats(const float* __restrict__ x, float* __restrict__ mean,
               float* __restrict__ rstd, float* __restrict__ stdv)
{
  const int p = blockIdx.x * 256 + threadIdx.x;   // [0, NB*NL)
  const int b = p >> 11;
  const int l = p & (NL - 1);
  const float* xp = x + (size_t)b * NC * NL + l;
  float s = 0.f, s2 = 0.f;
  for (int c = 0; c < NC; ++c) {
    float v = xp[(size_t)c * NL];                 // coalesced across threads
    s += v; s2 += v * v;
  }
  const float m   = s * (1.0f / (float)NC);
  const float var = (s2 - (float)NC * m * m) * (1.0f / (float)(NC - 1)) + 1e-5f;
  const float sd  = sqrtf(var);
  mean[p] = m; stdv[p] = sd; rstd[p] = 1.0f / sd;
}

// ---------------------------------------------------------------------------
// Pass B: normalize + forward FFT per row; accumulate sum_c |X[b,c,f]|
// ---------------------------------------------------------------------------
__global__ __launch_bounds__(256)
void fdm_fwd_spec(const float* __restrict__ x, const float* __restrict__ mean,
                  const float* __restrict__ rstd, float* __restrict__ gsum)
{
  __shared__ float re[NL], im[NL];
  __shared__ float Wr[256], Wi[256], Wn[256];
  const int tid = threadIdx.x;
  const int row = blockIdx.x;                     // b*NC + c
  const int b   = row >> 9;
  dft_tables_init(Wr, Wi, Wn, tid);
  const float* xr = x    + (size_t)row * NL;
  const float* mb = mean + (size_t)b * NL;
  const float* rb = rstd + (size_t)b * NL;
  #pragma unroll
  for (int q = 0; q < 8; ++q) {
    const int l = tid + 256 * q;
    re[l] = (xr[l] - mb[l]) * rb[l];
    im[l] = 0.0f;
  }
  __syncthreads();
  fft2048(re, im, Wr, Wi, Wn, tid);
  for (int f = tid; f < NF; f += 256) {
    const float mag = sqrtf(re[f] * re[f] + im[f] * im[f]);
    atomicAdd(&gsum[b * NF + f], mag);
  }
}

// ---------------------------------------------------------------------------
// Pass C: boundaries -> bins, gating softmax, expand per-bin weight w[b,f]
// ---------------------------------------------------------------------------
__global__ __launch_bounds__(256)
void fdm_gating(const float* __restrict__ bb, const float* __restrict__ gw,
                const float* __restrict__ gb, const float* __restrict__ gsum,
                float* __restrict__ wfreq)
{
  const int b   = blockIdx.x;
  const int tid = threadIdx.x;
  __shared__ float part[NE][257];
  __shared__ float g[NE];

  // sorted sigmoid boundaries -> integer bin edges (redundant per thread)
  float sbd[NE - 1];
  for (int i = 0; i < NE - 1; ++i) sbd[i] = 1.0f / (1.0f + __expf(-bb[i]));
  for (int i = 1; i < NE - 1; ++i) {
    float key = sbd[i]; int j = i - 1;
    while (j >= 0 && sbd[j] > key) { sbd[j + 1] = sbd[j]; --j; }
    sbd[j + 1] = key;
  }
  int idx[NE + 1];
  idx[0] = 0; idx[NE] = NF;
  for (int i = 0; i < NE - 1; ++i) idx[i + 1] = (int)(sbd[i] * (float)NF);

  // logits[e] = (1/C) * sum_f gsum[b,f] * gate_w[e,f]
  float p[NE];
  #pragma unroll
  for (int e = 0; e < NE; ++e) p[e] = 0.f;
  for (int f = tid; f < NF; f += 256) {
    const float gi = gsum[b * NF + f] * (1.0f / (float)NC);
    #pragma unroll
    for (int e = 0; e < NE; ++e) p[e] += gi * gw[e * NF + f];
  }
  #pragma unroll
  for (int e = 0; e < NE; ++e) part[e][tid] = p[e];
  __syncthreads();
  for (int off = 128; off > 0; off >>= 1) {
    if (tid < off) {
      #pragma unroll
      for (int e = 0; e < NE; ++e) part[e][tid] += part[e][tid + off];
    }
    __syncthreads();
  }
  if (tid == 0) {
    float lg[NE], mx = -1e30f;
    for (int e = 0; e < NE; ++e) { lg[e] = part[e][0] + gb[e]; mx = fmaxf(mx, lg[e]); }
    float den = 0.f;
    for (int e = 0; e < NE; ++e) { lg[e] = __expf(lg[e] - mx); den += lg[e]; }
    for (int e = 0; e < NE; ++e) g[e] = lg[e] / den;
  }
  __syncthreads();
  for (int f = tid; f < NF; f += 256) {
    float w = 0.f;
    #pragma unroll
    for (int e = 0; e < NE; ++e) w += (f >= idx[e] && f < idx[e + 1]) ? g[e] : 0.f;
    wfreq[b * NF + f] = w;
  }
}

// ---------------------------------------------------------------------------
// Pass D: normalize + FFT + gate*conj + FFT (== N*IFFT via conj) + denorm
// ---------------------------------------------------------------------------
__global__ __launch_bounds__(256)
void fdm_apply(const float* __restrict__ x, const float* __restrict__ mean,
               const float* __restrict__ rstd, const float* __restrict__ stdv,
               const float* __restrict__ wfreq, float* __restrict__ out)
{
  __shared__ float re[NL], im[NL];
  __shared__ float Wr[256], Wi[256], Wn[256];
  const int tid = threadIdx.x;
  const int row = blockIdx.x;                     // b*NC + c
  const int b   = row >> 9;
  dft_tables_init(Wr, Wi, Wn, tid);
  const float* xr = x     + (size_t)row * NL;
  const float* mb = mean  + (size_t)b * NL;
  const float* rb = rstd  + (size_t)b * NL;
  const float* sb = stdv  + (size_t)b * NL;
  const float* wb = wfreq + (size_t)b * NF;
  #pragma unroll
  for (int q = 0; q < 8; ++q) {
    const int l = tid + 256 * q;
    re[l] = (xr[l] - mb[l]) * rb[l];
    im[l] = 0.0f;
  }
  __syncthreads();
  fft2048(re, im, Wr, Wi, Wn, tid);
  // per-bin gate (symmetric extension keeps Hermitian symmetry) + conjugate
  #pragma unroll
  for (int q = 0; q < 8; ++q) {
    const int k  = tid + 256 * q;
    const int fs = (k <= NL / 2) ? k : (NL - k);
    const float s = wb[fs];
    const float a = re[k] * s;
    const float c = -im[k] * s;
    re[k] = a; im[k] = c;
  }
  __syncthreads();
  fft2048(re, im, Wr, Wi, Wn, tid);   // conj(FFT(conj(X)))/N == IFFT; take Re
  const float inv = 1.0f / (float)NL;
  float* o = out + (size_t)row * NL;
  #pragma unroll
  for (int q = 0; q < 8; ++q) {
    const int l = tid + 256 * q;
    o[l] = re[l] * inv * sb[l] + mb[l];
  }
}

// ---------------------------------------------------------------------------
extern "C" void kernel_launch(void* const* d_in, const int* in_sizes, int n_in,
                              void* d_out, int out_size, void* d_ws, size_t ws_size,
                              hipStream_t stream)
{
  (void)in_sizes; (void)n_in; (void)out_size; (void)ws_size;
  const float* x  = (const float*)d_in[0];   // [B, C, L]
  const float* bb = (const float*)d_in[1];   // [E-1]
  const float* gw = (const float*)d_in[2];   // [E, F]
  const float* gb = (const float*)d_in[3];   // [E]
  float* out = (float*)d_out;                // [B, C, L]

  float* ws    = (float*)d_ws;
  float* mean  = ws;                          // NB*NL
  float* rstd  = mean + NB * NL;              // NB*NL
  float* stdv  = rstd + NB * NL;              // NB*NL
  float* gsum  = stdv + NB * NL;              // NB*NF
  float* wfreq = gsum + NB * NF;              // NB*NF
  // total ws use: (3*65536 + 2*32800)*4 B ~= 1.0 MB

  hipMemsetAsync(gsum, 0, (size_t)NB * NF * sizeof(float), stream);
  fdm_stats   <<<(NB * NL) / 256, 256, 0, stream>>>(x, mean, rstd, stdv);
  fdm_fwd_spec<<<NB * NC,         256, 0, stream>>>(x, mean, rstd, gsum);
  fdm_gating  <<<NB,              256, 0, stream>>>(bb, gw, gb, gsum, wfreq);
  fdm_apply   <<<NB * NC,         256, 0, stream>>>(x, mean, rstd, stdv, wfreq, out);
}